// ACCGraphSAGE_81329500717447
// MI455X (gfx1250) — compile-verified
//
#include <hip/hip_runtime.h>

typedef __attribute__((ext_vector_type(2))) float v2f;
typedef __attribute__((ext_vector_type(8))) float v8f;

#define HID 128

// ---------------------------------------------------------------------------
// Degree accumulation: deg[dst[e]] += 1.0f
// ---------------------------------------------------------------------------
__global__ void deg_kernel(const int* __restrict__ dst, float* __restrict__ deg, int nE) {
    int e = blockIdx.x * blockDim.x + threadIdx.x;
    if (e < nE) atomicAdd(&deg[dst[e]], 1.0f);
}

// inv_deg[i] = 1 / max(deg[i], 1)   (in place)
__global__ void invdeg_kernel(float* __restrict__ deg, int n) {
    int i = blockIdx.x * blockDim.x + threadIdx.x;
    if (i < n) {
        float d = deg[i];
        deg[i] = 1.0f / fmaxf(d, 1.0f);
    }
}

// ---------------------------------------------------------------------------
// Edge gather/scatter: agg[dst[e], :] += h[src[e], colOff:colOff+128]
// h is read from the concatenated output buffer (row stride = outStride).
// One wave (32 lanes) per edge, 4 floats per lane.
// ---------------------------------------------------------------------------
__global__ __launch_bounds__(256)
void edge_scatter_kernel(const int* __restrict__ src, const int* __restrict__ dst,
                         const float* __restrict__ hbase, int outStride, int colOff,
                         float* __restrict__ agg, int nE) {
    int wid  = (blockIdx.x * blockDim.x + threadIdx.x) >> 5;
    int lane = threadIdx.x & 31;
    if (wid >= nE) return;
    int s = src[wid];
    int d = dst[wid];
    const float4 v = ((const float4*)(hbase + (size_t)s * outStride + colOff))[lane];
    float* ap = agg + (size_t)d * HID + lane * 4;
    atomicAdd(ap + 0, v.x);
    atomicAdd(ap + 1, v.y);
    atomicAdd(ap + 2, v.z);
    atomicAdd(ap + 3, v.w);
}

// ---------------------------------------------------------------------------
// Fused (optional inv_deg scale) + GEMM + ReLU using V_WMMA_F32_16X16X4_F32.
//   out[row, colOff + c] = relu( sum_k A[row,k]*scale(row) * W[k,c] )
// Block = 256 threads = 8 waves. Block owns 16 rows; wave w owns columns
// [16w, 16w+16). A tile (16x128 f32, 8 KB) staged in LDS, scaled during
// staging. K-loop: 32 x wmma_f32_16x16x4_f32.
//
// VGPR layouts (CDNA5 ISA 7.12.2):
//   A 16x4 f32 : lane m (m=lane&15), hi=lane>>4 ; VGPR0 = A[m][k+2*hi],
//                VGPR1 = A[m][k+2*hi+1]  -> contiguous pair, 8B-aligned LDS read
//   B 4x16 f32 : col n = lane&15 ; VGPR0 = B[2*hi][n], VGPR1 = B[2*hi+1][n]
//   C/D 16x16  : lane l, VGPR v -> D[v + 8*(l>>4)][l&15]
// ---------------------------------------------------------------------------
__global__ __launch_bounds__(256)
void gemm_relu_wmma(const float* __restrict__ A, int aStride,
                    const float* __restrict__ W,
                    const float* __restrict__ invdeg,   // nullptr => no scaling
                    float* __restrict__ out, int outStride, int colOff,
                    int nRows) {
    __shared__ float sA[16 * HID];

    const int tid   = threadIdx.x;
    const int node0 = blockIdx.x * 16;

    // ---- stage 16x128 A tile into LDS (256 threads x 8 floats) ----
    {
        int r   = tid >> 4;          // 0..15 tile row
        int c   = (tid & 15) * 8;    // 0,8,...,120
        int row = node0 + r;
        float scale = 1.0f;
        float4 x0 = make_float4(0.f, 0.f, 0.f, 0.f);
        float4 x1 = make_float4(0.f, 0.f, 0.f, 0.f);
        if (row < nRows) {
            const float4* ap = (const float4*)(A + (size_t)row * aStride + c);
            x0 = ap[0];
            x1 = ap[1];
            if (invdeg) scale = invdeg[row];
        }
        float* sp = sA + r * HID + c;
        sp[0] = x0.x * scale; sp[1] = x0.y * scale;
        sp[2] = x0.z * scale; sp[3] = x0.w * scale;
        sp[4] = x1.x * scale; sp[5] = x1.y * scale;
        sp[6] = x1.z * scale; sp[7] = x1.w * scale;
    }
    __syncthreads();

    const int lane    = tid & 31;
    const int hi      = lane >> 4;       // 0 or 1
    const int m       = lane & 15;       // A row / B col within tile
    const int colTile = (tid >> 5) * 16; // wave id * 16

    v8f acc = {};
    const float* wc = W + colTile + m;   // column (colTile+m) of W, stride HID

    #pragma unroll
    for (int k = 0; k < HID; k += 4) {
        v2f a = *(const v2f*)(sA + m * HID + k + 2 * hi);
        v2f b;
        b.x = wc[(k + 2 * hi) * HID];
        b.y = wc[(k + 2 * hi + 1) * HID];
        acc = __builtin_amdgcn_wmma_f32_16x16x4_f32(
            /*neg_a=*/false, a, /*neg_b=*/false, b,
            /*c_mod=*/(short)0, acc, /*reuse_a=*/false, /*reuse_b=*/false);
    }

    // ---- store D with ReLU ----
    #pragma unroll
    for (int v = 0; v < 8; ++v) {
        int row = node0 + v + 8 * hi;
        if (row < nRows)
            out[(size_t)row * outStride + colOff + colTile + m] = fmaxf(acc[v], 0.0f);
    }
}

// ---------------------------------------------------------------------------
// Launch
// Inputs: feat[N,128] f32, src[E] i32, dst[E] i32, W0[128,128] f32,
//         Ws[L,128,128] f32.  Output: [N, (L+1)*128] f32.
// Workspace: agg (N*128 f32) + deg (N f32)  ~= 25.8 MB.
// ---------------------------------------------------------------------------
extern "C" void kernel_launch(void* const* d_in, const int* in_sizes, int n_in,
                              void* d_out, int out_size, void* d_ws, size_t ws_size,
                              hipStream_t stream) {
    (void)n_in; (void)out_size; (void)ws_size;

    const float* feat = (const float*)d_in[0];
    const int*   src  = (const int*)d_in[1];
    const int*   dst  = (const int*)d_in[2];
    const float* W0   = (const float*)d_in[3];
    const float* Ws   = (const float*)d_in[4];

    const int N       = in_sizes[0] / HID;
    const int nE      = in_sizes[1];
    const int nLayers = in_sizes[4] / (HID * HID);
    const int outW    = (nLayers + 1) * HID;   // 512

    float* out = (float*)d_out;
    float* agg = (float*)d_ws;
    float* deg = agg + (size_t)N * HID;

    // degrees -> inverse degrees
    hipMemsetAsync(deg, 0, (size_t)N * sizeof(float), stream);
    deg_kernel<<<(nE + 255) / 256, 256, 0, stream>>>(dst, deg, nE);
    invdeg_kernel<<<(N + 255) / 256, 256, 0, stream>>>(deg, N);

    const int gBlocks = (N + 15) / 16;

    // layer 0: h0 = relu(feat @ W0) -> out[:, 0:128]
    gemm_relu_wmma<<<gBlocks, 256, 0, stream>>>(
        feat, HID, W0, nullptr, out, outW, 0, N);

    for (int l = 0; l < nLayers; ++l) {
        hipMemsetAsync(agg, 0, (size_t)N * HID * sizeof(float), stream);
        edge_scatter_kernel<<<(nE + 7) / 8, 256, 0, stream>>>(
            src, dst, out, outW, l * HID, agg, nE);
        // h_{l+1} = relu((agg * inv_deg) @ Ws[l]) -> out[:, (l+1)*128 : (l+2)*128]
        gemm_relu_wmma<<<gBlocks, 256, 0, stream>>>(
            agg, HID, Ws + (size_t)l * HID * HID, deg, out, outW, (l + 1) * HID, N);
    }
}